// SprecherMultiLayerNetwork_90048284328333
// MI455X (gfx1250) — compile-verified
//
#include <hip/hip_runtime.h>
#include <hip/hip_bf16.h>
#include <math.h>

// ---------------------------------------------------------------------------
// Sprecher multi-layer network, MI455X (gfx1250).
//
// Roofline: 3 * B*D*D = 402M spline evals, ~2 MB HBM traffic per layer
// => purely VALU + LDS-gather bound; HBM (23.3 TB/s) is irrelevant.
// The contraction sum_i lambda_i * phi(h[b,i] + eta*q[o]) has the
// nonlinearity inside a 3-index tensor -> no A(b,k)*B(k,o) factorization,
// so WMMA cannot host it (the MAC is 2 of ~9 ops per element anyway).
// CDNA5-specific paths used instead:
//   - GLOBAL_LOAD_ASYNC_TO_LDS_B128 + s_wait_asynccnt for h-tile staging
//   - v_med3_f32 / v_fract_f32 to minimize per-eval VALU ops
//   - LDS tables stored as (c, c[k+1]-c[k]) pairs: one ds_read_b64 + one
//     v_fma per spline eval
// ---------------------------------------------------------------------------

#define DD       512
#define BB       512
#define LLAYERS  3
#define BT       16          // b-tile
#define OT       32          // o-tile (2 outputs per thread)
#define ROWP     516         // padded LDS row stride (floats): breaks bank conflicts
#define NTHREADS 256
#define UMAX     98.999992f  // largest float < 99: (int) of it is 98

#if defined(__has_builtin)
#  if __has_builtin(__builtin_amdgcn_global_load_async_to_lds_b128)
#    define ATHENA_HAVE_ASYNC 1
#  endif
#endif
#ifndef ATHENA_HAVE_ASYNC
#  define ATHENA_HAVE_ASYNC 0
#endif

#if ATHENA_HAVE_ASYNC
// Toolchain signature (probe-confirmed round 1/2): (AS1 v4i*, AS3 v4i*, Iimm, Iimm)
typedef __attribute__((ext_vector_type(4))) int athena_v4i;
typedef __attribute__((address_space(1))) athena_v4i athena_gv4i;
typedef __attribute__((address_space(3))) athena_v4i athena_lv4i;
#endif

// ---------------------------------------------------------------------------
// Prep: build per-layer spline tables in workspace.
//   phiT[l][k] = (phi_coef[k], phi_coef[k+1]-phi_coef[k]),  k in [0,99)
//   PhiT[l][k] = normalized Phi coeff pairs (affine normalization folded in)
// ---------------------------------------------------------------------------
__global__ __launch_bounds__(128)
void sprecher_prep_kernel(const float* __restrict__ phi_log_inc,
                          const float* __restrict__ Phi_coeffs,
                          float2* __restrict__ phiT,
                          float2* __restrict__ PhiT)
{
    __shared__ float tmp[100];
    __shared__ float coef[100];
    const int t = threadIdx.x;

    for (int l = 0; l < LLAYERS; ++l) {
        // ---- monotonic phi: softplus -> cumsum -> normalize ----
        if (t < 100) {
            float x = phi_log_inc[l * 100 + t];
            tmp[t] = (x > 20.f) ? x : log1pf(expf(x));
        }
        __syncthreads();
        if (t == 0) {
            float c = 0.f;
            for (int k = 0; k < 100; ++k) { c += tmp[k]; coef[k] = c; }
            float inv = 1.f / (c + 1e-8f);
            for (int k = 0; k < 100; ++k) coef[k] *= inv;
        }
        __syncthreads();
        if (t < 99)
            phiT[l * 100 + t] = make_float2(coef[t], coef[t + 1] - coef[t]);
        __syncthreads();

        // ---- Phi: pre-normalize by (min,max) of coeffs ----
        if (t < 100) tmp[t] = Phi_coeffs[l * 100 + t];
        __syncthreads();
        if (t == 0) {
            float mn = tmp[0], mx = tmp[0];
            for (int k = 1; k < 100; ++k) {
                mn = fminf(mn, tmp[k]);
                mx = fmaxf(mx, tmp[k]);
            }
            float inv = 1.f / (mx - mn + 1e-8f);
            for (int k = 0; k < 100; ++k) coef[k] = (tmp[k] - mn) * inv;
        }
        __syncthreads();
        if (t < 99)
            PhiT[l * 100 + t] = make_float2(coef[t], coef[t + 1] - coef[t]);
        __syncthreads();
    }
}

// ---------------------------------------------------------------------------
// Main layer kernel: one 16(b) x 32(o) tile per 256-thread block;
// each thread produces two o-outputs sharing the h/lambda LDS reads.
// ---------------------------------------------------------------------------
__global__ __launch_bounds__(NTHREADS)
void sprecher_layer_kernel(const float* __restrict__ hin,
                           float*       __restrict__ hout,
                           const float* __restrict__ lam,     // lambdas[l]
                           const float2* __restrict__ phiTab, // phiT + l*100
                           const float2* __restrict__ PhiTab, // PhiT + l*100
                           const float* __restrict__ eta_p,
                           const float* __restrict__ dc_p,
                           const float* __restrict__ dr_p,
                           const float* __restrict__ cc_p,
                           const float* __restrict__ cr_p,
                           const float* __restrict__ resw_p)
{
    __shared__ __align__(16) float  sh[BT * ROWP];   // raw h tile (padded rows)
    __shared__ __align__(16) float  slam[DD];
    __shared__ float2 sphi[100];
    __shared__ float2 sPhi[100];

    const int tid = threadIdx.x;
    const int b0  = blockIdx.x * BT;
    const int o0  = blockIdx.y * OT;

    for (int k = tid; k < 99; k += NTHREADS) { sphi[k] = phiTab[k]; sPhi[k] = PhiTab[k]; }
    for (int i = tid; i < DD; i += NTHREADS) slam[i] = lam[i];

    // ---- h tile: 16 rows x 512 floats -> LDS via CDNA5 async engine ----
#if ATHENA_HAVE_ASYNC
    #pragma unroll
    for (int j = 0; j < (BT * DD / 4) / NTHREADS; ++j) {   // 8 x b128 per thread
        int idx4 = tid + j * NTHREADS;                     // 0..2047
        int row  = idx4 >> 7;                              // /128 vec4 per row
        int col  = (idx4 & 127) << 2;
        __builtin_amdgcn_global_load_async_to_lds_b128(
            (athena_gv4i*)(hin + (size_t)(b0 + row) * DD + col),
            (athena_lv4i*)(sh + row * ROWP + col),
            0, 0);
    }
#  if defined(__has_builtin) && __has_builtin(__builtin_amdgcn_s_wait_asynccnt)
    __builtin_amdgcn_s_wait_asynccnt(0);
#  else
    asm volatile("s_wait_asynccnt 0" ::: "memory");
#  endif
#else
    for (int j = 0; j < (BT * DD / 4) / NTHREADS; ++j) {
        int idx4 = tid + j * NTHREADS;
        int row  = idx4 >> 7;
        int col  = (idx4 & 127) << 2;
        float4 v = *(const float4*)(hin + (size_t)(b0 + row) * DD + col);
        *(float4*)(sh + row * ROWP + col) = v;
    }
#endif
    __syncthreads();

    // ---- per-thread scalars (uniform -> scalar regs) ----
    const float eta  = *eta_p;
    const float dc   = *dc_p, dr = *dr_p, cc = *cc_p, cr = *cr_p;
    const float rw   = *resw_p;
    const float in_min    = dc - dr;
    const float in_max    = dc + dr;
    const float out_min   = cc - cr;
    const float out_span  = 2.f * cr;
    const float invspan99 = 99.f / (2.f * dr);

    const int   ol  = tid & 15;
    const int   bl  = tid >> 4;
    const int   oa  = o0 + ol;          // first output column
    const int   ob  = o0 + ol + 16;     // second output column
    const float g0  = eta * 99.f * (float)oa;
    const float g1  = eta * 99.f * (float)ob;
    const float* hrow = sh + bl * ROWP;

    // u = med3(99h + 99*eta*q, 0, 98.999992); k = (int)u; t = fract(u)
    // pairs (c, dc) -> phi = fma(t, dc, c).  7 VALU + 1 ds_read_b64 per eval.
#define PHI_MAC2(hx, lx)                                                   \
    do {                                                                   \
        float u0 = __builtin_fmaf((hx), 99.f, g0);                         \
        float u1 = __builtin_fmaf((hx), 99.f, g1);                         \
        u0 = __builtin_amdgcn_fmed3f(u0, 0.f, UMAX);                       \
        u1 = __builtin_amdgcn_fmed3f(u1, 0.f, UMAX);                       \
        int   k0 = (int)u0,  k1 = (int)u1;                                 \
        float t0 = __builtin_amdgcn_fractf(u0);                            \
        float t1 = __builtin_amdgcn_fractf(u1);                            \
        float2 p0 = sphi[k0];                                              \
        float2 p1 = sphi[k1];                                              \
        acc0 = __builtin_fmaf((lx), __builtin_fmaf(t0, p0.y, p0.x), acc0); \
        acc1 = __builtin_fmaf((lx), __builtin_fmaf(t1, p1.y, p1.x), acc1); \
    } while (0)

    float acc0 = 0.f, acc1 = 0.f;
    #pragma unroll 4
    for (int i = 0; i < DD; i += 4) {
        const float4 hv = *(const float4*)(hrow + i);   // ds_read_b128 (2 rows/wave)
        const float4 lv = *(const float4*)(slam + i);   // ds_read_b128 (broadcast)
        PHI_MAC2(hv.x, lv.x);
        PHI_MAC2(hv.y, lv.y);
        PHI_MAC2(hv.z, lv.z);
        PHI_MAC2(hv.w, lv.w);
    }
#undef PHI_MAC2

    // ---- outer Phi spline (pre-normalized table) + residual ----
    auto epilogue = [&](float acc, int og) -> float {
        float s  = acc + (float)og;                          // + q[o]
        float sc = fminf(fmaxf(s, in_min), in_max);
        float v  = (sc - in_min) * invspan99;                // in [0,99]
        v = __builtin_amdgcn_fmed3f(v, 0.f, UMAX);
        int   k2 = (int)v;
        float t2 = __builtin_amdgcn_fractf(v);
        float2 P = sPhi[k2];
        float rawn = __builtin_amdgcn_fmed3f(__builtin_fmaf(t2, P.y, P.x), 0.f, 1.f);
        return __builtin_fmaf(out_span, rawn, out_min);
    };

    const float act0 = epilogue(acc0, oa) + rw * hrow[oa];
    const float act1 = epilogue(acc1, ob) + rw * hrow[ob];

    float* orow = hout + (size_t)(b0 + bl) * DD;
    orow[oa] = act0;
    orow[ob] = act1;
}

// ---------------------------------------------------------------------------
// Final: out[b] = output_scale * sum_o act[b, o]   (deterministic tree sum)
// ---------------------------------------------------------------------------
__global__ __launch_bounds__(256)
void sprecher_reduce_kernel(const float* __restrict__ act,
                            const float* __restrict__ oscale_p,
                            float* __restrict__ out)
{
    __shared__ float red[256];
    const int b = blockIdx.x;
    const int t = threadIdx.x;
    red[t] = act[(size_t)b * DD + t] + act[(size_t)b * DD + t + 256];
    __syncthreads();
    for (int sft = 128; sft > 0; sft >>= 1) {
        if (t < sft) red[t] += red[t + sft];
        __syncthreads();
    }
    if (t == 0) out[b] = red[0] * (*oscale_p);
}

// ---------------------------------------------------------------------------
extern "C" void kernel_launch(void* const* d_in, const int* in_sizes, int n_in,
                              void* d_out, int out_size, void* d_ws, size_t ws_size,
                              hipStream_t stream)
{
    (void)in_sizes; (void)n_in; (void)out_size; (void)ws_size;

    const float* x    = (const float*)d_in[0];   // [B, D]
    const float* lam  = (const float*)d_in[1];   // [L, D]
    const float* eta  = (const float*)d_in[2];   // [L]
    const float* pli  = (const float*)d_in[3];   // [L, 100]
    const float* Pco  = (const float*)d_in[4];   // [L, 100]
    const float* dc   = (const float*)d_in[5];
    const float* dr   = (const float*)d_in[6];
    const float* cc   = (const float*)d_in[7];
    const float* cr   = (const float*)d_in[8];
    const float* rw   = (const float*)d_in[9];   // [L]
    const float* osc  = (const float*)d_in[10];

    float*  ws   = (float*)d_ws;
    float*  buf0 = ws;                                   // [B, D]
    float*  buf1 = ws + (size_t)BB * DD;                 // [B, D]
    float2* phiT = (float2*)(ws + 2 * (size_t)BB * DD);  // [L, 100] pairs
    float2* PhiT = phiT + LLAYERS * 100;                 // [L, 100] pairs

    sprecher_prep_kernel<<<1, 128, 0, stream>>>(pli, Pco, phiT, PhiT);

    const dim3 grid(BB / BT, DD / OT);                   // (32, 16)
    // layer 0: x -> buf0
    sprecher_layer_kernel<<<grid, NTHREADS, 0, stream>>>(
        x,    buf0, lam + 0 * DD, phiT + 0,   PhiT + 0,   eta + 0, dc, dr, cc, cr, rw + 0);
    // layer 1: buf0 -> buf1
    sprecher_layer_kernel<<<grid, NTHREADS, 0, stream>>>(
        buf0, buf1, lam + 1 * DD, phiT + 100, PhiT + 100, eta + 1, dc, dr, cc, cr, rw + 1);
    // layer 2: buf1 -> buf0 (pre-sum activations)
    sprecher_layer_kernel<<<grid, NTHREADS, 0, stream>>>(
        buf1, buf0, lam + 2 * DD, phiT + 200, PhiT + 200, eta + 2, dc, dr, cc, cr, rw + 2);
    // final sum over o, scaled
    sprecher_reduce_kernel<<<BB, 256, 0, stream>>>(buf0, osc, (float*)d_out);
}